// ResNet_24249385353635
// MI455X (gfx1250) — compile-verified
//
#include <hip/hip_runtime.h>
#include <math.h>

// RetinaNet post-processing for MI455X (gfx1250, wave32).
//  A = 200000 anchors, C = 80 classes, MAX_DET = 100 keeps/class.
//  Pipeline: decode+clip -> threshold+transpose (ASYNCcnt per-lane async LDS
//  staging) -> per-class greedy NMS (fused suppress+argmax sweep, TDM
//  tensor_load_to_lds double-buffered box tiles, TENSORcnt) -> gather.
//
// Workspace layout:
//   boxes   : A*4 floats            (3.2 MB)
//   scoresT : C*A floats            (64  MB)  mutable per-class score rows
//   idxbuf  : C*MAX_DET ints
//   valbuf  : C*MAX_DET ints
// ~67.3 MB total; scoresT + boxes are L2-resident (192 MB), so the 100 NMS
// rounds stream from L2, not HBM.

#define MAXDETS   100
#define SCORE_TH  0.05f
#define IOU_TH    0.5f
#define NMS_T     1024          // 32 waves (wave32) per workgroup
#define TILE      1024          // boxes per TDM-staged LDS tile (16 KB/buffer)

// ---------------------------------------------------------------------------
// 1) Box decode + clip (elementwise, float4 streams)
// ---------------------------------------------------------------------------
__global__ void decode_kernel(const float* __restrict__ anchors,
                              const float* __restrict__ regression,
                              const int* __restrict__ img_h,
                              const int* __restrict__ img_w,
                              float* __restrict__ boxes, int A)
{
    int a = blockIdx.x * blockDim.x + threadIdx.x;
    if (a >= A) return;
    float4 an = ((const float4*)anchors)[a];
    float4 rg = ((const float4*)regression)[a];
    float W = (float)img_w[0], H = (float)img_h[0];

    float wa  = an.z - an.x;
    float ha  = an.w - an.y;
    float cxa = an.x + 0.5f * wa;
    float cya = an.y + 0.5f * ha;

    float cx = cxa + rg.x * 0.1f * wa;
    float cy = cya + rg.y * 0.1f * ha;
    float w  = expf(rg.z * 0.2f) * wa;
    float h  = expf(rg.w * 0.2f) * ha;

    float4 b;
    b.x = fminf(fmaxf(cx - 0.5f * w, 0.0f), W);
    b.y = fminf(fmaxf(cy - 0.5f * h, 0.0f), H);
    b.z = fminf(fmaxf(cx + 0.5f * w, 0.0f), W);
    b.w = fminf(fmaxf(cy + 0.5f * h, 0.0f), H);
    ((float4*)boxes)[a] = b;
}

// ---------------------------------------------------------------------------
// Per-lane async global->LDS copy (GLOBAL_LOAD_ASYNC_TO_LDS_B128, ASYNCcnt)
// ---------------------------------------------------------------------------
__device__ __forceinline__ void async_copy_b128(unsigned lds_byte_off,
                                                const float4* gaddr)
{
    asm volatile("global_load_async_to_lds_b128 %0, %1, off"
                 :: "v"(lds_byte_off), "v"(gaddr) : "memory");
}

// ---------------------------------------------------------------------------
// 2) Threshold + transpose: scoresT[c*A + a] = cls[a*C + c] > TH ? v : -1
//    Input tile (32 anchors x C classes) is contiguous in global memory, so
//    it is staged into LDS with per-lane async b128 copies.
// ---------------------------------------------------------------------------
__global__ void thresh_transpose_kernel(const float* __restrict__ cls,
                                        float* __restrict__ scoresT,
                                        int A, int C)
{
    __shared__ __align__(16) float tile[32 * 128];   // supports C <= 128
    const int abase = blockIdx.x * 32;
    const int tid   = threadIdx.x;
    const int total = 32 * C;

    if (abase + 32 <= A) {
        // fast path: contiguous 32*C floats -> async b128 burst into LDS
        const float4* src = (const float4*)(cls + (size_t)abase * C);
        const int nvec = total >> 2;
        for (int e = tid; e < nvec; e += blockDim.x)
            async_copy_b128((unsigned)(size_t)&tile[4 * e], src + e);
        asm volatile("s_wait_asynccnt 0x0" ::: "memory");
    } else {
        for (int e = tid; e < total; e += blockDim.x) {
            int al = e / C, cc = e - al * C;
            int a  = abase + al;
            tile[e] = (a < A) ? cls[(size_t)a * C + cc] : -1.0f;
        }
    }
    __syncthreads();

    for (int e = tid; e < total; e += blockDim.x) {
        int cc = e >> 5, al = e & 31;          // 32 consecutive anchors -> coalesced
        int a  = abase + al;
        if (a < A) {
            float v = tile[al * C + cc];
            scoresT[(size_t)cc * A + a] = (v > SCORE_TH) ? v : -1.0f;
        }
    }
}

// ---------------------------------------------------------------------------
// TDM: issue a 1-D tensor_load_to_lds descriptor (issued by one wave; the DMA
// engine moves the tile while all 32 waves compute). Tracked by TENSORcnt.
//   n_rem  : elements (4 B) remaining from gsrc to end of buffer (OOB -> 0)
//   n_tile : elements (4 B) in the tile (TILE boxes * 4)
// ---------------------------------------------------------------------------
typedef unsigned u32x4 __attribute__((ext_vector_type(4)));
typedef unsigned u32x8 __attribute__((ext_vector_type(8)));

__device__ __forceinline__ void tdm_load_tile(const float* gsrc,
                                              unsigned lds_byte_off,
                                              unsigned n_rem, unsigned n_tile)
{
    unsigned long long ga = (unsigned long long)(size_t)gsrc;
    u32x4 g0;
    g0[0] = 1u;                                   // count=1, user descriptor
    g0[1] = lds_byte_off;                         // lds_addr
    g0[2] = (unsigned)ga;                         // global_addr[31:0]
    g0[3] = (unsigned)((ga >> 32) & 0x01FFFFFFu)  // global_addr[56:32]
          | (2u << 30);                           // type = 2 ("image")
    u32x8 g1;
    g1[0] = (2u << 16);                           // wg_mask=0, data_size=4B
    g1[1] = (n_rem & 0xFFFFu) << 16;              // tensor_dim0[15:0]
    g1[2] = (n_rem >> 16);                        // tensor_dim0[31:16]
    g1[3] = (n_tile & 0xFFFFu) << 16;             // tile_dim0
    g1[4] = 0; g1[5] = 0; g1[6] = 0; g1[7] = 0;   // 1-D: no dim1/2, no strides
    asm volatile("tensor_load_to_lds %0, %1" :: "s"(g0), "s"(g1) : "memory");
}

// ---------------------------------------------------------------------------
// 3) Greedy NMS, one workgroup per class. Each round fuses "suppress by the
//    previous winner" with "argmax of updated scores" into one sweep. Box
//    tiles are DMA'd by the TDM into a double-buffered LDS tile.
// ---------------------------------------------------------------------------
__global__ __launch_bounds__(NMS_T)
void nms_kernel(const float* __restrict__ boxes,
                float* __restrict__ scoresT,
                int* __restrict__ idxbuf,
                int* __restrict__ valbuf,
                int A)
{
    const int c    = blockIdx.x;
    const int tid  = threadIdx.x;
    const int wave = tid >> 5;
    const int lane = tid & 31;
    float* sc = scoresT + (size_t)c * A;
    const float4* bx = (const float4*)boxes;

    __shared__ __align__(16) float4 tile[2][TILE];   // 32 KB double buffer
    __shared__ float  rv[32];
    __shared__ int    ri[32];
    __shared__ float4 s_pbox;
    __shared__ float  s_parea;
    __shared__ int    s_pidx;
    __shared__ int    s_pvalid;

    if (tid == 0) {
        s_pidx = -1; s_pvalid = 0;
        s_pbox = make_float4(0.f, 0.f, 0.f, 0.f);
        s_parea = 0.f;
    }
    __syncthreads();

    const int ntiles = (A + TILE - 1) / TILE;

    for (int k = 0; k < MAXDETS; ++k) {
        const float4 pbox  = s_pbox;
        const float  parea = s_parea;
        const int    pidx  = s_pidx;
        const int    pval  = s_pvalid;
        __syncthreads();

        // prologue: wave 0 launches the DMA for tile 0 into buffer 0
        if (wave == 0)
            tdm_load_tile(boxes, (unsigned)(size_t)&tile[0][0],
                          (unsigned)A * 4u, TILE * 4u);

        float bestv = -__builtin_inff();
        int   besti = 0x7fffffff;
        int   buf   = 0;

        for (int t = 0; t < ntiles; ++t, buf ^= 1) {
            const int base  = t * TILE;
            const int nbase = base + TILE;
            if (wave == 0) {
                if (nbase < A) {
                    tdm_load_tile(boxes + (size_t)nbase * 4,
                                  (unsigned)(size_t)&tile[buf ^ 1][0],
                                  (unsigned)(A - nbase) * 4u, TILE * 4u);
                    // keep 1 descriptor in flight; current tile is complete
                    asm volatile("s_wait_tensorcnt 0x1" ::: "memory");
                } else {
                    asm volatile("s_wait_tensorcnt 0x0" ::: "memory");
                }
            }
            __syncthreads();                   // current tile visible to all waves

            const int a = base + tid;
            if (a < A) {
                float s = sc[a];
                if (a == pidx) {               // reference: s.at[i].set(-1)
                    if (s != -1.0f) sc[a] = -1.0f;
                    s = -1.0f;
                } else if (pval && s > 0.0f) {
                    float4 b = tile[buf][tid]; // ds_load_b128 from DMA'd tile
                    float xx1 = fmaxf(pbox.x, b.x);
                    float yy1 = fmaxf(pbox.y, b.y);
                    float xx2 = fminf(pbox.z, b.z);
                    float yy2 = fminf(pbox.w, b.w);
                    float inter = fmaxf(xx2 - xx1, 0.0f) * fmaxf(yy2 - yy1, 0.0f);
                    float area  = (b.z - b.x) * (b.w - b.y);
                    float iou   = inter / (parea + area - inter);
                    if (iou > IOU_TH) { sc[a] = -1.0f; s = -1.0f; }
                }
                if (s > bestv) { bestv = s; besti = a; }   // ascending a -> first max
            }
            __syncthreads();                   // tile[buf] safe to re-stage
        }

        // wave32 argmax: shfl reduction in-wave, then one 32-entry LDS stage.
        // Lowest-index tie-break matches jnp.argmax.
        for (int off = 16; off > 0; off >>= 1) {
            float ov = __shfl_down(bestv, off, 32);
            int   oi = __shfl_down(besti, off, 32);
            if (ov > bestv || (ov == bestv && oi < besti)) { bestv = ov; besti = oi; }
        }
        if (lane == 0) { rv[wave] = bestv; ri[wave] = besti; }
        __syncthreads();
        if (wave == 0) {
            bestv = rv[lane]; besti = ri[lane];
            for (int off = 16; off > 0; off >>= 1) {
                float ov = __shfl_down(bestv, off, 32);
                int   oi = __shfl_down(besti, off, 32);
                if (ov > bestv || (ov == bestv && oi < besti)) { bestv = ov; besti = oi; }
            }
            if (lane == 0) {
                int i = besti;
                if (i == 0x7fffffff) i = 0;    // degenerate: all suppressed
                int valid = (bestv > 0.0f) ? 1 : 0;
                idxbuf[c * MAXDETS + k] = i;
                valbuf[c * MAXDETS + k] = valid;
                float4 b = bx[i];
                s_pbox   = b;
                s_parea  = (b.z - b.x) * (b.w - b.y);
                s_pidx   = i;
                s_pvalid = valid;
            }
        }
        __syncthreads();
    }
}

// ---------------------------------------------------------------------------
// 4) Gather: d_out = [sel_scores (C*M) | class_ids-as-float (C*M) | boxes (C*M*4)]
// ---------------------------------------------------------------------------
__global__ void gather_kernel(const float* __restrict__ cls,
                              const float* __restrict__ boxes,
                              const int* __restrict__ idxbuf,
                              const int* __restrict__ valbuf,
                              float* __restrict__ out, int C)
{
    int t  = blockIdx.x * blockDim.x + threadIdx.x;
    int CM = C * MAXDETS;
    if (t >= CM) return;
    int c = t / MAXDETS;
    int i = idxbuf[t];
    int v = valbuf[t];
    float f = v ? 1.0f : 0.0f;

    out[t]      = cls[(size_t)i * C + c] * f;
    out[CM + t] = v ? (float)c : -1.0f;
    float4 b = ((const float4*)boxes)[i];
    float* ob = out + 2 * CM + 4 * t;
    ob[0] = b.x * f; ob[1] = b.y * f; ob[2] = b.z * f; ob[3] = b.w * f;
}

// ---------------------------------------------------------------------------
extern "C" void kernel_launch(void* const* d_in, const int* in_sizes, int n_in,
                              void* d_out, int out_size, void* d_ws, size_t ws_size,
                              hipStream_t stream)
{
    const float* cls = (const float*)d_in[0];   // [A,C]
    const float* reg = (const float*)d_in[1];   // [A,4]
    const float* anc = (const float*)d_in[2];   // [A,4]
    const int*   ih  = (const int*)d_in[3];
    const int*   iw  = (const int*)d_in[4];

    const int A = in_sizes[1] / 4;
    const int C = in_sizes[0] / A;

    float* boxes   = (float*)d_ws;                      // A*4 floats
    float* scoresT = boxes + (size_t)A * 4;             // C*A floats
    int*   idxbuf  = (int*)(scoresT + (size_t)C * A);   // C*MAXDETS
    int*   valbuf  = idxbuf + C * MAXDETS;              // C*MAXDETS

    decode_kernel<<<(A + 255) / 256, 256, 0, stream>>>(anc, reg, ih, iw, boxes, A);
    thresh_transpose_kernel<<<(A + 31) / 32, 256, 0, stream>>>(cls, scoresT, A, C);
    nms_kernel<<<C, NMS_T, 0, stream>>>(boxes, scoresT, idxbuf, valbuf, A);
    gather_kernel<<<(C * MAXDETS + 255) / 256, 256, 0, stream>>>(
        cls, boxes, idxbuf, valbuf, (float*)d_out, C);
}